// Seq2Seq_20822001451262
// MI455X (gfx1250) — compile-verified
//
#include <hip/hip_runtime.h>
#include <hip/hip_bf16.h>
#include <math.h>

// Problem constants (match reference setup_inputs)
#define HID    256
#define GDIM   1024      // 4*HID
#define NB     512       // BATCH
#define KBEAM  4
#define MBEAM  2048      // NB*KBEAM
#define TMAX   32
#define OUTV   4

typedef float v2f __attribute__((ext_vector_type(2)));
typedef float v8f __attribute__((ext_vector_type(8)));

__device__ __forceinline__ float sigf(float x) { return 1.0f / (1.0f + expf(-x)); }

// ---------------------------------------------------------------------------
// g = x @ Wih^T + h @ Whh^T + bih + bhh      (M x 1024), K = 256
// One wave computes a 16x64 strip of g (4 n-tiles) with V_WMMA_F32_16X16X4_F32.
// A fragments (x-row and h-row pairs) are loaded once per K-step and reused
// against 4 weight tiles; each tile's x and h WMMAs run back to back on one accumulator.
// If tok != nullptr, x is an embedding table and row m reads emb[tok[m%nTok]].
// ---------------------------------------------------------------------------
__global__ __launch_bounds__(128)
void lstm_gemm_wmma(const float* __restrict__ xa,   // (M,256) activations OR (VOCAB,256) emb
                    const int*   __restrict__ tok,  // nullptr => xa is (M,256)
                    int nTok,
                    const float* __restrict__ h,    // (M,256)
                    const float* __restrict__ Wih,  // (1024,256)
                    const float* __restrict__ Whh,  // (1024,256)
                    const float* __restrict__ bih,  // (1024)
                    const float* __restrict__ bhh,  // (1024)
                    float* __restrict__ g,          // (M,1024)
                    int M)
{
    const int NTB  = 4;                 // n-tiles per wave
    const int NGRP = GDIM / (16 * NTB); // 16 strips along N
    int wave = (blockIdx.x * blockDim.x + threadIdx.x) >> 5;
    int lane = threadIdx.x & 31;
    int mt  = wave / NGRP;
    int ng  = wave - mt * NGRP;
    int nt0 = ng * NTB;
    int lr = lane & 15;     // row (A) / col (B) within tile
    int lh = lane >> 4;     // half-wave selects K pair

    int m = mt * 16 + lr;
    long xrow = tok ? (long)tok[m % nTok] : (long)m;
    const float* xptr = xa  + xrow * HID + 2 * lh;
    const float* hptr = h   + (long)m * HID + 2 * lh;
    const float* wip  = Wih + (long)(nt0 * 16 + lr) * HID + 2 * lh;
    const float* whp  = Whh + (long)(nt0 * 16 + lr) * HID + 2 * lh;

    // Hint the weight strips into cache (global_prefetch_b8; no counter traffic)
    __builtin_prefetch(wip, 0, 1);
    __builtin_prefetch(whp, 0, 1);

    v8f acc[NTB];
    #pragma unroll
    for (int j = 0; j < NTB; ++j) acc[j] = (v8f){0.f,0.f,0.f,0.f,0.f,0.f,0.f,0.f};

    #pragma unroll 4
    for (int kb = 0; kb < HID; kb += 4) {
        v2f ax = *(const v2f*)(xptr + kb);
        v2f ah = *(const v2f*)(hptr + kb);
        #pragma unroll
        for (int j = 0; j < NTB; ++j) {
            // weight tile j is a compile-time 16*HID-float offset (16 KB)
            v2f bi = *(const v2f*)(wip + (long)j * 16 * HID + kb);
            acc[j] = __builtin_amdgcn_wmma_f32_16x16x4_f32(false, ax, false, bi,
                                                           (short)0, acc[j], false, false);
            v2f bh = *(const v2f*)(whp + (long)j * 16 * HID + kb);
            acc[j] = __builtin_amdgcn_wmma_f32_16x16x4_f32(false, ah, false, bh,
                                                           (short)0, acc[j], false, false);
        }
    }

    long mbase = (long)(mt * 16 + 8 * lh) * GDIM;
    #pragma unroll
    for (int j = 0; j < NTB; ++j) {
        int n = (nt0 + j) * 16 + lr;
        float bias = bih[n] + bhh[n];
        #pragma unroll
        for (int r = 0; r < 8; ++r)
            g[mbase + (long)r * GDIM + n] = acc[j][r] + bias;
    }
}

// ---------------------------------------------------------------------------
// LSTM gate nonlinearity: cn = sig(f)*c + sig(i)*tanh(g); hn = sig(o)*tanh(cn)
// ---------------------------------------------------------------------------
__global__ __launch_bounds__(256)
void lstm_gates(const float* __restrict__ g, const float* __restrict__ cprev,
                float* __restrict__ cnew, float* __restrict__ hnew,
                float* __restrict__ xnext, int M)
{
    int i = blockIdx.x * blockDim.x + threadIdx.x;
    if (i >= M * HID) return;
    int m = i >> 8, j = i & 255;
    const float* gr = g + (long)m * GDIM;
    float ig = gr[j], fg = gr[HID + j], gg = gr[2 * HID + j], og = gr[3 * HID + j];
    float c  = sigf(fg) * cprev[i] + sigf(ig) * tanhf(gg);
    float hh = sigf(og) * tanhf(c);
    cnew[i] = c; hnew[i] = hh; xnext[i] = hh;
}

// out = log_softmax(top @ fcW^T + fcb), V=4
__global__ __launch_bounds__(256)
void fc_logsoftmax(const float* __restrict__ top, const float* __restrict__ fcW,
                   const float* __restrict__ fcb, float* __restrict__ out, int M)
{
    int r = blockIdx.x * blockDim.x + threadIdx.x;
    if (r >= M) return;
    float a0 = 0.f, a1 = 0.f, a2 = 0.f, a3 = 0.f;
    const float* t = top + (long)r * HID;
    for (int k = 0; k < HID; ++k) {
        float tv = t[k];
        a0 = fmaf(tv, fcW[k],           a0);
        a1 = fmaf(tv, fcW[HID + k],     a1);
        a2 = fmaf(tv, fcW[2 * HID + k], a2);
        a3 = fmaf(tv, fcW[3 * HID + k], a3);
    }
    a0 += fcb[0]; a1 += fcb[1]; a2 += fcb[2]; a3 += fcb[3];
    float mx = fmaxf(fmaxf(a0, a1), fmaxf(a2, a3));
    float s  = expf(a0 - mx) + expf(a1 - mx) + expf(a2 - mx) + expf(a3 - mx);
    float l  = mx + logf(s);
    float* o = out + (long)r * OUTV;
    o[0] = a0 - l; o[1] = a1 - l; o[2] = a2 - l; o[3] = a3 - l;
}

__global__ __launch_bounds__(256)
void zero_f32(float* __restrict__ p, int n)
{
    int i = blockIdx.x * blockDim.x + threadIdx.x;
    if (i < n) p[i] = 0.f;
}

// Tile encoder state (2,512,256) -> decoder state (2,2048,256): row k*B+b <- b
__global__ __launch_bounds__(256)
void tile_state(const float* __restrict__ he, const float* __restrict__ ce,
                float* __restrict__ hd, float* __restrict__ cd)
{
    int i = blockIdx.x * blockDim.x + threadIdx.x;       // 2*2048*256
    if (i >= 2 * MBEAM * HID) return;
    int l   = i / (MBEAM * HID);
    int rem = i - l * (MBEAM * HID);
    int r = rem >> 8, e = rem & 255;
    int b = r % NB;
    long s = (long)l * NB * HID + (long)b * HID + e;
    hd[i] = he[s]; cd[i] = ce[s];
}

// outputs[0] = out0 ; cum = max_v out0
__global__ __launch_bounds__(256)
void step0_init(const float* __restrict__ out, float* __restrict__ O,
                float* __restrict__ cum, int M)
{
    int r = blockIdx.x * blockDim.x + threadIdx.x;
    if (r >= M) return;
    float v0 = out[r * 4], v1 = out[r * 4 + 1], v2 = out[r * 4 + 2], v3 = out[r * 4 + 3];
    O[r * 4] = v0; O[r * 4 + 1] = v1; O[r * 4 + 2] = v2; O[r * 4 + 3] = v3;
    cum[r] = fmaxf(fmaxf(v0, v1), fmaxf(v2, v3));
}

// Exact top-4 of 16 with lax.top_k tie semantics (lowest index first).
// Length-penalty division by lp>0 is monotone -> skipped.
__global__ __launch_bounds__(256)
void beam_topk(const float* __restrict__ cum, const float* __restrict__ out,
               float* __restrict__ cumNew, int* __restrict__ rows)
{
    int b = blockIdx.x * blockDim.x + threadIdx.x;
    if (b >= NB) return;
    float p[16];
    #pragma unroll
    for (int k = 0; k < KBEAM; ++k) {
        int rr = k * NB + b;
        float cb = cum[rr];
        #pragma unroll
        for (int v = 0; v < OUTV; ++v) p[k * OUTV + v] = cb + out[rr * 4 + v];
    }
    bool used[16];
    #pragma unroll
    for (int i = 0; i < 16; ++i) used[i] = false;
    for (int kk = 0; kk < KBEAM; ++kk) {
        int bi = 0; float best = -__builtin_inff();
        for (int i = 0; i < 16; ++i)
            if (!used[i] && p[i] > best) { best = p[i]; bi = i; }
        used[bi] = true;
        int beam = bi >> 2;
        rows[kk * NB + b]   = beam * NB + b;
        cumNew[kk * NB + b] = p[bi];      // == cum[rows] + out[rows, tks]
    }
}

// {hd,cd}[l,r,:] = {hs,cs}[l, rows[r], :]   over (2, 2048, 256)
__global__ __launch_bounds__(256)
void gather_hc(const float* __restrict__ hs, const float* __restrict__ cs,
               float* __restrict__ hd, float* __restrict__ cd,
               const int* __restrict__ rows)
{
    int i = blockIdx.x * blockDim.x + threadIdx.x;
    if (i >= 2 * MBEAM * HID) return;
    int l   = i / (MBEAM * HID);
    int rem = i - l * (MBEAM * HID);
    int r = rem >> 8, e = rem & 255;
    long s = (long)l * MBEAM * HID + (long)rows[r] * HID + e;
    hd[i] = hs[s];
    cd[i] = cs[s];
}

// Odst[t,r,:] = (t==tcur) ? outcur[rows[r],:] : Osrc[t, rows[r], :]
__global__ __launch_bounds__(256)
void gather_outputs(const float* __restrict__ Osrc, float* __restrict__ Odst,
                    const float* __restrict__ outcur, const int* __restrict__ rows,
                    int tcur)
{
    int i = blockIdx.x * blockDim.x + threadIdx.x;      // 32*2048*4
    if (i >= TMAX * MBEAM * OUTV) return;
    int t   = i / (MBEAM * OUTV);
    int rem = i - t * (MBEAM * OUTV);
    int r = rem >> 2, v = rem & 3;
    int sr = rows[r];
    Odst[i] = (t == tcur) ? outcur[sr * 4 + v] : Osrc[(long)t * MBEAM * OUTV + sr * 4 + v];
}

// d_out[t,b,v] = O[t, b, v] for b < 512
__global__ __launch_bounds__(256)
void write_out(const float* __restrict__ O, float* __restrict__ dout)
{
    int i = blockIdx.x * blockDim.x + threadIdx.x;      // 32*512*4
    if (i >= TMAX * NB * OUTV) return;
    int t   = i / (NB * OUTV);
    int rem = i - t * (NB * OUTV);
    int b = rem >> 2, v = rem & 3;
    dout[i] = O[(long)t * MBEAM * OUTV + (long)b * OUTV + v];
}

// ---------------------------------------------------------------------------
extern "C" void kernel_launch(void* const* d_in, const int* in_sizes, int n_in,
                              void* d_out, int out_size, void* d_ws, size_t ws_size,
                              hipStream_t stream)
{
    const int*   src     = (const int*)  d_in[0];   // (1,32,512)
    const float* enc_emb = (const float*)d_in[2];
    const float* enc_Wih = (const float*)d_in[3];   // (2,1024,256)
    const float* enc_Whh = (const float*)d_in[4];
    const float* enc_bih = (const float*)d_in[5];   // (2,1024)
    const float* enc_bhh = (const float*)d_in[6];
    const float* dec_emb = (const float*)d_in[7];
    const float* dec_Wih = (const float*)d_in[8];
    const float* dec_Whh = (const float*)d_in[9];
    const float* dec_bih = (const float*)d_in[10];
    const float* dec_bhh = (const float*)d_in[11];
    const float* fcW     = (const float*)d_in[12];  // (4,256)
    const float* fcb     = (const float*)d_in[13];  // (4)

    const long WSLICE = (long)GDIM * HID;   // per-layer weight slice (1024*256)
    const long BSLICE = GDIM;               // per-layer bias slice

    // ---- workspace carve-up (floats) ----
    float* ws   = (float*)d_ws;
    float* xbuf = ws;                               // 2048*256
    float* gbuf = xbuf + (long)MBEAM * HID;         // 2048*1024
    float* henc = gbuf + (long)MBEAM * GDIM;        // 2*512*256
    float* cenc = henc + 2L * NB * HID;
    float* hA   = cenc + 2L * NB * HID;             // 2*2048*256
    float* hB   = hA   + 2L * MBEAM * HID;
    float* cA   = hB   + 2L * MBEAM * HID;
    float* cB   = cA   + 2L * MBEAM * HID;
    float* OA   = cB   + 2L * MBEAM * HID;          // 32*2048*4
    float* OB   = OA   + (long)TMAX * MBEAM * OUTV;
    float* outc = OB   + (long)TMAX * MBEAM * OUTV; // 2048*4
    float* cumA = outc + (long)MBEAM * OUTV;        // 2048
    float* cumB = cumA + MBEAM;
    int*   rows = (int*)(cumB + MBEAM);             // 2048 ints

    const int GEMM_BLK = 128;   // 4 waves/block; waves = M -> blocks = M/4
    const int TPB = 256;

    // ---- encoder: zero state, 31 reversed steps ----
    zero_f32<<<(2 * 2 * NB * HID + TPB - 1) / TPB, TPB, 0, stream>>>(henc, 2 * 2 * NB * HID);

    for (int i = 0; i < 31; ++i) {
        const int* tokt = src + (31 - i) * NB;
        // layer 0 (embedding fused into A-load)
        lstm_gemm_wmma<<<NB / 4, GEMM_BLK, 0, stream>>>(enc_emb, tokt, NB,
            henc, enc_Wih, enc_Whh, enc_bih, enc_bhh, gbuf, NB);
        lstm_gates<<<NB, TPB, 0, stream>>>(gbuf, cenc, cenc, henc, xbuf, NB);
        // layer 1
        lstm_gemm_wmma<<<NB / 4, GEMM_BLK, 0, stream>>>(xbuf, (const int*)nullptr, 1,
            henc + (long)NB * HID, enc_Wih + WSLICE, enc_Whh + WSLICE,
            enc_bih + BSLICE, enc_bhh + BSLICE, gbuf, NB);
        lstm_gates<<<NB, TPB, 0, stream>>>(gbuf, cenc + (long)NB * HID,
            cenc + (long)NB * HID, henc + (long)NB * HID, xbuf, NB);
    }

    // ---- decoder: tile state across beams ----
    tile_state<<<(2 * MBEAM * HID + TPB - 1) / TPB, TPB, 0, stream>>>(henc, cenc, hA, cA);

    // step 0: tok = tile(src[0,0,:], K)
    {
        const int* tok0 = src;   // t = 0
        lstm_gemm_wmma<<<MBEAM / 4, GEMM_BLK, 0, stream>>>(dec_emb, tok0, NB,
            hA, dec_Wih, dec_Whh, dec_bih, dec_bhh, gbuf, MBEAM);
        lstm_gates<<<MBEAM, TPB, 0, stream>>>(gbuf, cA, cA, hA, xbuf, MBEAM);
        lstm_gemm_wmma<<<MBEAM / 4, GEMM_BLK, 0, stream>>>(xbuf, (const int*)nullptr, 1,
            hA + (long)MBEAM * HID, dec_Wih + WSLICE, dec_Whh + WSLICE,
            dec_bih + BSLICE, dec_bhh + BSLICE, gbuf, MBEAM);
        lstm_gates<<<MBEAM, TPB, 0, stream>>>(gbuf, cA + (long)MBEAM * HID,
            cA + (long)MBEAM * HID, hA + (long)MBEAM * HID, xbuf, MBEAM);
        fc_logsoftmax<<<MBEAM / TPB, TPB, 0, stream>>>(xbuf, fcW, fcb, outc, MBEAM);
        step0_init<<<MBEAM / TPB, TPB, 0, stream>>>(outc, OA, cumA, MBEAM);
    }

    float *hCur = hA, *hNxt = hB, *cCur = cA, *cNxt = cB;
    float *Ocur = OA, *Onxt = OB, *cumCur = cumA, *cumNxt = cumB;

    for (int t = 1; t < TMAX; ++t) {
        const int* tokt = src + t * NB;
        lstm_gemm_wmma<<<MBEAM / 4, GEMM_BLK, 0, stream>>>(dec_emb, tokt, NB,
            hCur, dec_Wih, dec_Whh, dec_bih, dec_bhh, gbuf, MBEAM);
        lstm_gates<<<MBEAM, TPB, 0, stream>>>(gbuf, cCur, cCur, hCur, xbuf, MBEAM);
        lstm_gemm_wmma<<<MBEAM / 4, GEMM_BLK, 0, stream>>>(xbuf, (const int*)nullptr, 1,
            hCur + (long)MBEAM * HID, dec_Wih + WSLICE, dec_Whh + WSLICE,
            dec_bih + BSLICE, dec_bhh + BSLICE, gbuf, MBEAM);
        lstm_gates<<<MBEAM, TPB, 0, stream>>>(gbuf, cCur + (long)MBEAM * HID,
            cCur + (long)MBEAM * HID, hCur + (long)MBEAM * HID, xbuf, MBEAM);
        fc_logsoftmax<<<MBEAM / TPB, TPB, 0, stream>>>(xbuf, fcW, fcb, outc, MBEAM);

        beam_topk<<<(NB + TPB - 1) / TPB, TPB, 0, stream>>>(cumCur, outc, cumNxt, rows);
        gather_hc<<<(2 * MBEAM * HID + TPB - 1) / TPB, TPB, 0, stream>>>(
            hCur, cCur, hNxt, cNxt, rows);
        gather_outputs<<<(TMAX * MBEAM * OUTV + TPB - 1) / TPB, TPB, 0, stream>>>(
            Ocur, Onxt, outc, rows, t);

        float* tmp;
        tmp = hCur; hCur = hNxt; hNxt = tmp;
        tmp = cCur; cCur = cNxt; cNxt = tmp;
        tmp = Ocur; Ocur = Onxt; Onxt = tmp;
        tmp = cumCur; cumCur = cumNxt; cumNxt = tmp;
    }

    write_out<<<(TMAX * NB * OUTV + TPB - 1) / TPB, TPB, 0, stream>>>(Ocur, (float*)d_out);
}